// GCN2MLP_45509473469000
// MI455X (gfx1250) — compile-verified
//
#include <hip/hip_runtime.h>

typedef __attribute__((ext_vector_type(2))) float v2f;
typedef __attribute__((ext_vector_type(8))) float v8f;

enum { EPI_SCALE = 0, EPI_BIAS_RELU = 1, EPI_BIAS = 2 };

// ---------------------------------------------------------------------------
// Degree accumulation: deg[dst] += 1 for each real edge (self loop added later)
// ---------------------------------------------------------------------------
__global__ __launch_bounds__(256) void deg_kernel(const int* __restrict__ dst,
                                                  float* __restrict__ deg, int nedges) {
  int tid = blockIdx.x * blockDim.x + threadIdx.x;
  if (tid < nedges) atomicAdd(&deg[dst[tid]], 1.0f);
}

// dinv[i] = rsqrt(deg[i] + 1)   (+1 = self loop; deg >= 1 always -> no zero case)
__global__ __launch_bounds__(256) void dinv_kernel(float* __restrict__ deg_dinv, int n) {
  int i = blockIdx.x * blockDim.x + threadIdx.x;
  if (i < n) deg_dinv[i] = rsqrtf(deg_dinv[i] + 1.0f);
}

// ---------------------------------------------------------------------------
// Tiled fp32 GEMM using V_WMMA_F32_16X16X4_F32.
//   C(nrows x DOUT) = A(nrows x DIN) @ W(DIN x DOUT), fused epilogue:
//     EPI_SCALE:     C[r][c] = acc * dinv[r]        (GCN pre-aggregation scale)
//     EPI_BIAS_RELU: C[r][c] = max(acc + bias[c],0) (MLP hidden)
//     EPI_BIAS:      C[r][c] = acc + bias[c]        (MLP out)
// Block = 256 threads = 8 waves; each wave owns one 16x16 output tile.
// W column tile (DIN x 16) staged in LDS in [K][16] layout (bank-conflict free).
// ---------------------------------------------------------------------------
template<int DIN, int DOUT, int MODE>
__global__ __launch_bounds__(256) void gemm_wmma_kernel(
    const float* __restrict__ A, const float* __restrict__ W,
    const float* __restrict__ bias, const float* __restrict__ dinv,
    float* __restrict__ C, int nrows)
{
  __shared__ float wlds[DIN * 16];
  const int colBase = blockIdx.y * 16;
  for (int i = threadIdx.x; i < DIN * 16; i += 256)
    wlds[i] = W[(i >> 4) * DOUT + colBase + (i & 15)];
  __syncthreads();

  const int lane    = threadIdx.x & 31;
  const int wave    = threadIdx.x >> 5;
  const int rowTile = blockIdx.x * 8 + wave;
  if (rowTile * 16 >= nrows) return;   // uniform per wave: EXEC stays all-1s

  const int m       = lane & 15;       // col-within-tile for B/C/D; row for A
  const int hi      = lane >> 4;       // lane-group select (K+2 / M+8 halves)
  const int rowBase = rowTile * 16;
  const int col     = colBase + m;

  // A fragment source: lane m holds A[rowBase+m][k + 2*hi .. +1] (aligned float2)
  const float* arow = A + (size_t)(rowBase + m) * DIN + 2 * hi;

  v8f acc = {0.f, 0.f, 0.f, 0.f, 0.f, 0.f, 0.f, 0.f};
#pragma unroll 8
  for (int k = 0; k < DIN; k += 4) {
    v2f a = *(const v2f*)(arow + k);
    v2f b;
    b.x = wlds[(k + 2 * hi) * 16 + m];       // B row K   = k + 2*hi
    b.y = wlds[(k + 2 * hi + 1) * 16 + m];   // B row K+1
    acc = __builtin_amdgcn_wmma_f32_16x16x4_f32(
        /*neg_a=*/false, a, /*neg_b=*/false, b,
        /*c_mod=*/(short)0, acc, /*reuse_a=*/false, /*reuse_b=*/false);
  }

  float bval = 0.f;
  if (MODE != EPI_SCALE) bval = bias[col];

#pragma unroll
  for (int r = 0; r < 8; ++r) {
    int   row = rowBase + r + 8 * hi;        // C/D layout: VGPR r -> M = r + 8*hi
    float v   = acc[r];
    if (MODE == EPI_SCALE)          v *= dinv[row];
    else if (MODE == EPI_BIAS_RELU) v = fmaxf(v + bval, 0.f);
    else                            v = v + bval;
    C[(size_t)row * DOUT + col] = v;
  }
}

// ---------------------------------------------------------------------------
// Edge aggregation: out[dst] += hs[src] (hs already row-scaled by dinv).
// One thread = one float4 chunk of one edge; consecutive threads cover
// consecutive features of the same edge -> coalesced gather, L2-resident atomics.
// ---------------------------------------------------------------------------
template<int D>
__global__ __launch_bounds__(256) void edge_agg_kernel(
    const float* __restrict__ hs, const int* __restrict__ src,
    const int* __restrict__ dst, float* __restrict__ out, int nedges)
{
  constexpr int QPE = D / 4;
  long long tid = (long long)blockIdx.x * blockDim.x + threadIdx.x;
  if (tid >= (long long)nedges * QPE) return;
  int e = (int)(tid / QPE);
  int q = (int)(tid % QPE);
  int s = src[e], d = dst[e];
  const float4 v = *(const float4*)(hs + (size_t)s * D + q * 4);
  float* o = out + (size_t)d * D + q * 4;
  atomicAdd(o + 0, v.x);
  atomicAdd(o + 1, v.y);
  atomicAdd(o + 2, v.z);
  atomicAdd(o + 3, v.w);
}

// ---------------------------------------------------------------------------
// GCN finalize: h[i][f] = relu(agg[i][f] * dinv[i] + bias[f])
// ---------------------------------------------------------------------------
template<int D>
__global__ __launch_bounds__(256) void gcn_post_kernel(
    const float* __restrict__ agg, const float* __restrict__ dinv,
    const float* __restrict__ bias, float* __restrict__ h, int n)
{
  constexpr int Q = D / 4;
  long long tid = (long long)blockIdx.x * blockDim.x + threadIdx.x;
  if (tid >= (long long)n * Q) return;
  int i  = (int)(tid / Q);
  int q  = (int)(tid % Q);
  float  s = dinv[i];
  float4 a = *(const float4*)(agg + (size_t)i * D + q * 4);
  float4 b = *(const float4*)(bias + q * 4);
  float4 r;
  r.x = fmaxf(a.x * s + b.x, 0.f);
  r.y = fmaxf(a.y * s + b.y, 0.f);
  r.z = fmaxf(a.z * s + b.z, 0.f);
  r.w = fmaxf(a.w * s + b.w, 0.f);
  *(float4*)(h + (size_t)i * D + q * 4) = r;
}

// ---------------------------------------------------------------------------
// Launch
// ---------------------------------------------------------------------------
extern "C" void kernel_launch(void* const* d_in, const int* in_sizes, int n_in,
                              void* d_out, int out_size, void* d_ws, size_t ws_size,
                              hipStream_t stream) {
  const float* x   = (const float*)d_in[0];
  const int*   ei  = (const int*)  d_in[1];
  const float* W1  = (const float*)d_in[2];
  const float* b1  = (const float*)d_in[3];
  const float* W2  = (const float*)d_in[4];
  const float* b2  = (const float*)d_in[5];
  const float* W3  = (const float*)d_in[6];
  const float* b3  = (const float*)d_in[7];
  const float* Wd1 = (const float*)d_in[8];
  const float* bd1 = (const float*)d_in[9];
  const float* Wd2 = (const float*)d_in[10];
  const float* bd2 = (const float*)d_in[11];
  const float* Wd3 = (const float*)d_in[12];
  const float* bd3 = (const float*)d_in[13];

  const int N = in_sizes[0] / 256;   // 50000 (== 3125 * 16)
  const int E = in_sizes[1] / 2;     // 800000
  const int* src = ei;
  const int* dst = ei + E;

  // Workspace: dinv (N floats) | buf1 (N x 128 floats)
  float* dinv = (float*)d_ws;
  float* buf1 = dinv + N;            // N*4 = 200000 B, 16B-aligned
  // d_out doubles as scratch mid-pipeline (fully overwritten by final GEMM)
  float* outf = (float*)d_out;

  const int RT = (N + 15) / 16;      // row tiles
  auto ggrid = [&](int dout) { return dim3((RT + 7) / 8, dout / 16); };
  auto eblocks = [&](int d) { return (int)(((long long)E * (d / 4) + 255) / 256); };
  auto pblocks = [&](int d) { return (int)(((long long)N * (d / 4) + 255) / 256); };

  // ---- degree / dinv ----
  hipMemsetAsync(dinv, 0, (size_t)N * sizeof(float), stream);
  deg_kernel<<<(E + 255) / 256, 256, 0, stream>>>(dst, dinv, E);
  dinv_kernel<<<(N + 255) / 256, 256, 0, stream>>>(dinv, N);

  // ---- GCN layer 1: 256 -> 128 ----
  gemm_wmma_kernel<256, 128, EPI_SCALE><<<ggrid(128), 256, 0, stream>>>(
      x, W1, nullptr, dinv, buf1, N);                                   // hs1 in buf1
  hipMemcpyAsync(outf, buf1, (size_t)N * 128 * sizeof(float),
                 hipMemcpyDeviceToDevice, stream);                      // agg1 = self loops
  edge_agg_kernel<128><<<eblocks(128), 256, 0, stream>>>(buf1, src, dst, outf, E);
  gcn_post_kernel<128><<<pblocks(128), 256, 0, stream>>>(outf, dinv, b1, buf1, N); // h1

  // ---- GCN layer 2: 128 -> 64 ----
  gemm_wmma_kernel<128, 64, EPI_SCALE><<<ggrid(64), 256, 0, stream>>>(
      buf1, W2, nullptr, dinv, outf, N);                                // hs2 in outf[0:N*64]
  hipMemcpyAsync(outf + (size_t)N * 64, outf, (size_t)N * 64 * sizeof(float),
                 hipMemcpyDeviceToDevice, stream);                      // agg2
  edge_agg_kernel<64><<<eblocks(64), 256, 0, stream>>>(outf, src, dst,
                                                       outf + (size_t)N * 64, E);
  gcn_post_kernel<64><<<pblocks(64), 256, 0, stream>>>(outf + (size_t)N * 64,
                                                       dinv, b2, buf1, N); // h2

  // ---- GCN layer 3: 64 -> 32 ----
  gemm_wmma_kernel<64, 32, EPI_SCALE><<<ggrid(32), 256, 0, stream>>>(
      buf1, W3, nullptr, dinv, outf, N);                                // hs3 in outf[0:N*32]
  hipMemcpyAsync(outf + (size_t)N * 32, outf, (size_t)N * 32 * sizeof(float),
                 hipMemcpyDeviceToDevice, stream);                      // agg3
  edge_agg_kernel<32><<<eblocks(32), 256, 0, stream>>>(outf, src, dst,
                                                       outf + (size_t)N * 32, E);
  gcn_post_kernel<32><<<pblocks(32), 256, 0, stream>>>(outf + (size_t)N * 32,
                                                       dinv, b3, buf1, N); // z in buf1

  // ---- MLP decoder: 32 -> 64 -> 128 -> 256 ----
  gemm_wmma_kernel<32, 64, EPI_BIAS_RELU><<<ggrid(64), 256, 0, stream>>>(
      buf1, Wd1, bd1, nullptr, outf, N);                                // d1 in outf
  gemm_wmma_kernel<64, 128, EPI_BIAS_RELU><<<ggrid(128), 256, 0, stream>>>(
      outf, Wd2, bd2, nullptr, buf1, N);                                // d2 in buf1
  gemm_wmma_kernel<128, 256, EPI_BIAS><<<ggrid(256), 256, 0, stream>>>(
      buf1, Wd3, bd3, nullptr, outf, N);                                // o -> d_out
}